// EarlyRewardLoss_29583734735591
// MI455X (gfx1250) — compile-verified
//
#include <hip/hip_runtime.h>
#include <hip/hip_bf16.h>
#include <math.h>

// Problem constants (fixed by the reference: T=128, N=4096, C=64)
#define T_DIM 128
#define N_DIM 4096
#define C_DIM 64
#define EPS_OVER_T 0.078125f   // EPSILON / T = 10 / 128
#define BLOCK 1024
#define ITEMS (N_DIM / BLOCK)  // 4 elements per thread
#define NWAVES (BLOCK / 32)    // 32 wave32s per workgroup

typedef __attribute__((ext_vector_type(4))) unsigned int tdm_u32x4;
typedef __attribute__((ext_vector_type(8))) int          tdm_i32x8;
typedef __attribute__((ext_vector_type(4))) int          tdm_i32x4;

// ---------------------------------------------------------------------------
// Kernel A: one workgroup per row t.
//   1) TDM-DMA the 16KB delta row into LDS (tensor_load_to_lds + s_wait_tensorcnt)
//   2) multiplicative prefix scan of (1 - delta) across n (budget)
//   3) fused Pt / gather / exp / dot-product partials, block-reduced
// ---------------------------------------------------------------------------
__global__ __launch_bounds__(BLOCK) void EarlyRewardLoss_row_kernel(
    const float* __restrict__ lp,     // (T, N, C) log-softmax
    const float* __restrict__ delta,  // (T, N) probability_stopping
    const int*   __restrict__ ytru,   // (N, T)
    float*       __restrict__ row_out // (T) per-row partials
) {
  __shared__ float sdelta[N_DIM];       // 16 KB staged row
  __shared__ float swp[NWAVES];         // per-wave chunk products
  __shared__ float sred[2 * NWAVES];    // final reduction buffers

  const int t    = blockIdx.x;
  const int tid  = threadIdx.x;
  const int lane = tid & 31;
  const int wave = tid >> 5;

  // ---- Stage delta row t into LDS via the Tensor Data Mover ----
#if defined(__gfx1250__) && __has_builtin(__builtin_amdgcn_tensor_load_to_lds)
  if (tid == 0) {
    unsigned long long gaddr =
        (unsigned long long)(const void*)(delta + (size_t)t * N_DIM);
    unsigned lds_off =
        (unsigned)(unsigned long long)(const void*)&sdelta[0]; // flat addr[31:0] == LDS byte offset

    tdm_u32x4 g0;
    g0[0] = 1u;                                         // count=1, user descriptor
    g0[1] = lds_off;                                    // lds_addr (bytes)
    g0[2] = (unsigned)(gaddr & 0xFFFFFFFFull);          // global_addr[31:0]
    g0[3] = (unsigned)((gaddr >> 32) & 0x01FFFFFFull)   // global_addr[56:32]
            | 0x80000000u;                              // type=2 ("image")

    tdm_i32x8 g1;
    g1[0] = (int)(2u << 16);                 // workgroup_mask=0, data_size=2 (4 bytes)
    g1[1] = (int)((unsigned)(N_DIM & 0xFFFF) << 16);     // tensor_dim0[15:0]
    g1[2] = (int)(((unsigned)N_DIM >> 16) | (1u << 16)); // tensor_dim0[31:16] | tensor_dim1[15:0]=1
    g1[3] = (int)((unsigned)N_DIM << 16);    // tensor_dim1[31:16]=0 | tile_dim0=4096
    g1[4] = 1;                               // tile_dim1=1, tile_dim2=0 (unused)
    g1[5] = N_DIM;                           // tensor_dim0_stride[31:0]
    g1[6] = 0;                               // stride0[47:32] | stride1[15:0]
    g1[7] = 0;                               // stride1[47:16]

    tdm_i32x4 g2 = {0, 0, 0, 0};             // dims 2/3 unused
    tdm_i32x4 g3 = {0, 0, 0, 0};             // dim 4 unused
    tdm_i32x8 g4 = {0, 0, 0, 0, 0, 0, 0, 0}; // extra operand (clang-23 lane), zero-filled

    __builtin_amdgcn_tensor_load_to_lds(g0, g1, g2, g3, g4, /*cpol=*/0);
    __builtin_amdgcn_s_wait_tensorcnt(0);
  }
  __syncthreads();
#else
  for (int i = tid; i < N_DIM; i += BLOCK)
    sdelta[i] = delta[(size_t)t * N_DIM + i];
  __syncthreads();
#endif

  // ---- Multiplicative scan: budget[n] = prod_{i<=n} f[i], f[0]=1, f[i]=1-delta[i] ----
  const int base = tid * ITEMS;
  float f[ITEMS];
  float cp = 1.0f;
#pragma unroll
  for (int i = 0; i < ITEMS; ++i) {
    const int n = base + i;
    const float fv = (n == 0) ? 1.0f : (1.0f - sdelta[n]);
    f[i] = fv;
    cp *= fv;
  }

  // wave-level inclusive product scan (wave32)
  float incl = cp;
#pragma unroll
  for (int off = 1; off < 32; off <<= 1) {
    const float v = __shfl_up(incl, off, 32);
    if (lane >= off) incl *= v;
  }
  if (lane == 31) swp[wave] = incl;
  __syncthreads();

  // cross-wave exclusive prefix (<=31 serial multiplies, deterministic)
  float wprefix = 1.0f;
  for (int w = 0; w < wave; ++w) wprefix *= swp[w];

  float excl = __shfl_up(incl, 1, 32);
  if (lane == 0) excl = 1.0f;
  float budget = wprefix * excl;   // product of all f strictly before this chunk

  // ---- Fused Pt / gathers / partial sums ----
  const float td = 1.0f - (float)t / (float)T_DIM;  // time decay for this row
  float accCls = 0.0f, accEr = 0.0f;
#pragma unroll
  for (int i = 0; i < ITEMS; ++i) {
    const int n = base + i;
    budget *= f[i];                                   // inclusive budget[n]
    const float pts = (n < N_DIM - 1) ? sdelta[n + 1] * budget : budget;
    const float Pt  = pts + EPS_OVER_T;

    // gathered[t, n] = lp[t, n, y_true[n, t]]
    const int   cls = ytru[(size_t)n * T_DIM + t];
    const float g   = lp[((size_t)t * N_DIM + n) * C_DIM + cls];

    // scrambled index for y_haty: k = n*128 + t -> t' = n>>5, n' = ((n&31)<<7) + t
    const int   t2   = n >> 5;
    const int   n2   = ((n & 31) << 7) + t;
    const int   cls2 = ytru[(size_t)n2 * T_DIM + t2];
    const float g2   = lp[((size_t)t2 * N_DIM + n2) * C_DIM + cls2];

    accCls += (-g) * Pt;          // cross-entropy * Pt
    accEr  += Pt * expf(g2);      // Pt * y_haty
  }

  // ---- deterministic block reduction ----
#pragma unroll
  for (int off = 16; off > 0; off >>= 1) {
    accCls += __shfl_down(accCls, off, 32);
    accEr  += __shfl_down(accEr, off, 32);
  }
  if (lane == 0) {
    sred[wave]          = accCls;
    sred[NWAVES + wave] = accEr;
  }
  __syncthreads();
  if (tid == 0) {
    float c = 0.0f, e = 0.0f;
    for (int w = 0; w < NWAVES; ++w) { c += sred[w]; e += sred[NWAVES + w]; }
    row_out[t] = c - td * e;     // CL_row - time_decay * ER_row
  }
}

// ---------------------------------------------------------------------------
// Kernel B: reduce 128 row partials -> scalar loss = (ALPHA/T)*sum
// ---------------------------------------------------------------------------
__global__ __launch_bounds__(T_DIM) void EarlyRewardLoss_final_kernel(
    const float* __restrict__ row_vals, float* __restrict__ out) {
  __shared__ float s[T_DIM / 32];
  const int tid = threadIdx.x;
  float v = row_vals[tid];
#pragma unroll
  for (int off = 16; off > 0; off >>= 1) v += __shfl_down(v, off, 32);
  if ((tid & 31) == 0) s[tid >> 5] = v;
  __syncthreads();
  if (tid == 0) {
    float tot = 0.0f;
#pragma unroll
    for (int w = 0; w < T_DIM / 32; ++w) tot += s[w];
    out[0] = (0.5f / (float)T_DIM) * tot;  // ALPHA*CL - (1-ALPHA)*ER, ALPHA=0.5
  }
}

extern "C" void kernel_launch(void* const* d_in, const int* in_sizes, int n_in,
                              void* d_out, int out_size, void* d_ws, size_t ws_size,
                              hipStream_t stream) {
  const float* lp    = (const float*)d_in[0];  // (T, N, C) fp32
  const float* ps    = (const float*)d_in[1];  // (T, N)    fp32
  const int*   ytru  = (const int*)d_in[2];    // (N, T)    int32
  float*       rows  = (float*)d_ws;           // 128 floats of scratch

  EarlyRewardLoss_row_kernel<<<T_DIM, BLOCK, 0, stream>>>(lp, ps, ytru, rows);
  EarlyRewardLoss_final_kernel<<<1, T_DIM, 0, stream>>>(rows, (float*)d_out);
}